// PPCALayer_57775900066620
// MI455X (gfx1250) — compile-verified
//
#include <hip/hip_runtime.h>
#include <hip/hip_bf16.h>
#include <stdint.h>

// Problem constants (from reference): x [64,256,56,56] f32, weight [3136,15] f32
#define B_      64
#define C_      256
#define HW_     3136
#define TS      112                 // spatial tile (divides 3136: 28 tiles/batch)
#define NTILES  (HW_ / TS)          // 28
#define THREADS 224                 // 7 waves of 32; (chunk g, pixel-quad q) = 8 x 28 work items

#if defined(__has_builtin)
#  if __has_builtin(__builtin_amdgcn_tensor_load_to_lds)
#    define HAVE_TDM 1
#  else
#    define HAVE_TDM 0
#  endif
#  if __has_builtin(__builtin_amdgcn_tensor_store_from_lds)
#    define HAVE_TDM_ST 1
#  else
#    define HAVE_TDM_ST 0
#  endif
#  if __has_builtin(__builtin_amdgcn_s_wait_tensorcnt)
#    define HAVE_TENSORCNT 1
#  else
#    define HAVE_TENSORCNT 0
#  endif
#else
#  define HAVE_TDM 0
#  define HAVE_TDM_ST 0
#  define HAVE_TENSORCNT 0
#endif

typedef unsigned int u32x4 __attribute__((ext_vector_type(4)));
typedef int          i32x4 __attribute__((ext_vector_type(4)));
typedef int          i32x8 __attribute__((ext_vector_type(8)));

__device__ __forceinline__ void tdm_wait0() {
#if HAVE_TENSORCNT
    __builtin_amdgcn_s_wait_tensorcnt(0);
#else
    asm volatile("s_wait_tensorcnt 0x0" ::: "memory");
#endif
}

// Build the 2-D Tensor DMA descriptor for a [C_ rows x TS cols] f32 tile of a
// row-stride-HW_ tensor, at global element address gaddr, landing at LDS byte
// offset laddr with packed row stride TS*4.
__device__ __forceinline__ void tdm_make_desc(uint64_t gaddr, uint32_t laddr,
                                              u32x4& g0, i32x8& g1) {
    g0[0] = 1u;                                       // count=1, user descriptor, no gather
    g0[1] = laddr;                                    // lds_addr   [63:32]
    g0[2] = (uint32_t)(gaddr & 0xFFFFFFFFu);          // global_addr[31:0]
    g0[3] = (uint32_t)((gaddr >> 32) & 0x01FFFFFFu)   // global_addr[56:32]
          | (2u << 30);                               // type = 2 ("image")

    g1[0] = (int)(2u << 16);                          // data_size = 2 (4 bytes)
    g1[1] = (int)((uint32_t)HW_ << 16);               // tensor_dim0[15:0]  -> bits[63:48]
    g1[2] = (int)(((uint32_t)HW_ >> 16)               // tensor_dim0[31:16]
          |       ((uint32_t)C_  << 16));             // tensor_dim1[15:0]
    g1[3] = (int)(((uint32_t)C_  >> 16)               // tensor_dim1[31:16]
          |       ((uint32_t)TS  << 16));             // tile_dim0 = 112
    g1[4] = (int)C_;                                  // tile_dim1 = 256, tile_dim2 = 0
    g1[5] = (int)HW_;                                 // tensor_dim0_stride low32 = 3136
    g1[6] = 0;                                        // stride hi16 | tensor_dim1_stride low16
    g1[7] = 0;
}

__global__ __launch_bounds__(THREADS)
void ppca_gate_kernel(const float* __restrict__ x,
                      const float* __restrict__ w,
                      float* __restrict__ out) {
    // Tile of x: all 256 channels x 112 pixels; channel-row stride = TS floats.
    __shared__ __align__(16) float tile[C_ * TS];   // 114688 B
    __shared__ __align__(16) float part[8 * TS];    // chunk partial sums, layout [g][pixel]
    __shared__ __align__(16) float gate[TS];

    const int blk = blockIdx.x;
    const int b   = blk / NTILES;
    const int t   = blk % NTILES;
    const int s0  = t * NTILES == 0 ? t * TS : t * TS;   // t*TS
    const uint64_t base = ((uint64_t)b * C_) * HW_ + (uint64_t)(t * TS);

    const int g = threadIdx.x / 28;          // channel chunk 0..7 (channels 32g..32g+31)
    const int q = threadIdx.x % 28;          // pixel quad 0..27  (pixels 4q..4q+3)

#if HAVE_TDM
    // ---- TDM load: one DMA of the whole [256 x 112] f32 tile into LDS.
    // Issued by wave 0 only (EXEC ignored; one issue per wave executing it).
    if (threadIdx.x < 32) {
        u32x4 g0; i32x8 g1;
        tdm_make_desc((uint64_t)(uintptr_t)(x + base),
                      (uint32_t)(uintptr_t)(&tile[0]), g0, g1);
        i32x4 gz4 = {0, 0, 0, 0};
        i32x8 gz8 = {0, 0, 0, 0, 0, 0, 0, 0};
        __builtin_amdgcn_tensor_load_to_lds(g0, g1, gz4, gz4, gz8, 0);
        tdm_wait0();
    }
    __syncthreads();
#else
    for (int i = threadIdx.x; i < C_ * TS / 4; i += THREADS) {
        const int c = i / (TS / 4);
        const int v = (i % (TS / 4)) * 4;
        const float4 d = *(const float4*)(x + base + (uint64_t)c * HW_ + v);
        *(float4*)&tile[c * TS + v] = d;
    }
    __syncthreads();
#endif

    // ---- Chunk reduction, vectorized: thread (g,q) sums 32 channels of its
    // chunk for 4 pixels with ds_load_b128 float4 accumulation.
    {
        float4 acc = make_float4(0.f, 0.f, 0.f, 0.f);
        const float* tp = &tile[(g * 32) * TS + q * 4];
#pragma unroll
        for (int c = 0; c < 32; ++c) {
            const float4 v = *(const float4*)(tp + c * TS);
            acc.x += v.x; acc.y += v.y; acc.z += v.z; acc.w += v.w;
        }
        *(float4*)&part[g * TS + q * 4] = acc;
    }
    __syncthreads();

    // ---- Per-pixel: 8 chunk sums -> 15 multi-scale features -> normalize ->
    //      dot with this pixel's weight row -> sigmoid gate.
    if (threadIdx.x < TS) {
        const int pos = threadIdx.x;
        float s8[8];
#pragma unroll
        for (int j = 0; j < 8; ++j) s8[j] = part[j * TS + pos];

        float f[15];
        float tot = 0.0f;
#pragma unroll
        for (int j = 0; j < 8; ++j) { f[7 + j] = s8[j] * (1.0f / 32.0f); tot += s8[j]; }
#pragma unroll
        for (int j = 0; j < 4; ++j) f[3 + j] = (s8[2 * j] + s8[2 * j + 1]) * (1.0f / 64.0f);
        f[1] = (s8[0] + s8[1] + s8[2] + s8[3]) * (1.0f / 128.0f);
        f[2] = (s8[4] + s8[5] + s8[6] + s8[7]) * (1.0f / 128.0f);
        f[0] = tot * (1.0f / 256.0f);

        float m = 0.0f, m2 = 0.0f;
#pragma unroll
        for (int j = 0; j < 15; ++j) { m += f[j]; m2 += f[j] * f[j]; }
        m  *= (1.0f / 15.0f);
        m2 *= (1.0f / 15.0f);
        const float var    = m2 - m * m;
        const float invstd = rsqrtf(fmaxf(var, 1e-30f));

        const float* wr = w + (uint64_t)(t * TS + pos) * 15;
        float z = 0.0f;
#pragma unroll
        for (int j = 0; j < 15; ++j) z += (f[j] - m) * invstd * wr[j];

        gate[pos] = 1.0f / (1.0f + __expf(-z));
    }
    __syncthreads();

    // ---- Scale pass: thread (g,q) scales its 32 channels x 4 pixels.
#if HAVE_TDM_ST
    {
        const float4 gv = *(const float4*)&gate[q * 4];
        float* tp = &tile[(g * 32) * TS + q * 4];
#pragma unroll
        for (int c = 0; c < 32; ++c) {
            float4 v = *(const float4*)(tp + c * TS);
            v.x *= gv.x; v.y *= gv.y; v.z *= gv.z; v.w *= gv.w;
            *(float4*)(tp + c * TS) = v;
        }
    }
    __syncthreads();
    // ---- TDM store: one DMA of the scaled [256 x 112] tile LDS -> out.
    if (threadIdx.x < 32) {
        u32x4 g0; i32x8 g1;
        tdm_make_desc((uint64_t)(uintptr_t)(out + base),
                      (uint32_t)(uintptr_t)(&tile[0]), g0, g1);
        i32x4 gz4 = {0, 0, 0, 0};
        i32x8 gz8 = {0, 0, 0, 0, 0, 0, 0, 0};
        __builtin_amdgcn_tensor_store_from_lds(g0, g1, gz4, gz4, gz8, 0);
        tdm_wait0();   // ensure DMA drains before LDS dealloc at wave end
    }
#else
    {
        const float4 gv = *(const float4*)&gate[q * 4];
        const float* tp = &tile[(g * 32) * TS + q * 4];
        float* op = out + base + (uint64_t)(g * 32) * HW_ + q * 4;
#pragma unroll
        for (int c = 0; c < 32; ++c) {
            float4 v = *(const float4*)(tp + c * TS);
            v.x *= gv.x; v.y *= gv.y; v.z *= gv.z; v.w *= gv.w;
            *(float4*)(op + (uint64_t)c * HW_) = v;
        }
    }
#endif
    (void)s0;
}

extern "C" void kernel_launch(void* const* d_in, const int* in_sizes, int n_in,
                              void* d_out, int out_size, void* d_ws, size_t ws_size,
                              hipStream_t stream) {
    (void)in_sizes; (void)n_in; (void)d_ws; (void)ws_size; (void)out_size;
    const float* x = (const float*)d_in[0];   // [64,256,56,56] f32
    const float* w = (const float*)d_in[1];   // [3136,15] f32
    float* out = (float*)d_out;               // [64,256,56,56] f32
    ppca_gate_kernel<<<dim3(B_ * NTILES), dim3(THREADS), 0, stream>>>(x, w, out);
}